// TrilinearInterpolation_7808250544317
// MI455X (gfx1250) — compile-verified
//
#include <hip/hip_runtime.h>
#include <math.h>

typedef float v4f __attribute__((ext_vector_type(4)));

typedef __attribute__((address_space(1))) int gas_int;  /* global addrspace */
typedef __attribute__((address_space(3))) int las_int;  /* LDS addrspace    */

#define LUT_DIM   33
#define LUT_N     (LUT_DIM * LUT_DIM * LUT_DIM)   /* 35937 entries            */
#define PLANE_PX  (1024 * 1024)                   /* pixels per channel plane */
#define GRP_BITS  18                              /* (1<<20)/4 groups/plane   */

/* ---- detect gfx1250 async global->LDS builtins (compile-probe guarded) ---- */
#if defined(__has_builtin)
#if __has_builtin(__builtin_amdgcn_global_load_async_to_lds_b32)
#define HAVE_ASYNC_LDS 1
#endif
#endif

/* =========================================================================
 * Kernel A: repack LUT from channel-major [3][35937] f32 into entry-major
 * float4 [35937] (x=ch0, y=ch1, z=ch2, w=pad) so one aligned b128 gather
 * fetches all three channels of a corner.  Uses the CDNA5 async global->LDS
 * path (ASYNCcnt) where it costs nothing: each lane stages its own 3 dwords
 * through LDS, waits on asynccnt, then emits one b128 store.
 * ========================================================================= */
__global__ __launch_bounds__(256)
void lut_pack_kernel(const float* __restrict__ lut, v4f* __restrict__ lut4)
{
    const int e = blockIdx.x * blockDim.x + threadIdx.x;
#if HAVE_ASYNC_LDS
    __shared__ float sbuf[3 * 256];
    const int t = threadIdx.x;
    if (e < LUT_N) {
#pragma unroll
        for (int c = 0; c < 3; ++c) {
            float* g = const_cast<float*>(lut) + c * LUT_N + e;
            float* l = &sbuf[c * 256 + t];
            __builtin_amdgcn_global_load_async_to_lds_b32(
                (gas_int*)g, (las_int*)l, /*offset=*/0, /*cpol=*/0);
        }
    }
#if __has_builtin(__builtin_amdgcn_s_wait_asynccnt)
    __builtin_amdgcn_s_wait_asynccnt(0);
#else
    asm volatile("s_wait_asynccnt 0" ::: "memory");
#endif
    if (e < LUT_N) {
        v4f v;
        v.x = sbuf[t];
        v.y = sbuf[256 + t];
        v.z = sbuf[512 + t];
        v.w = 0.0f;
        lut4[e] = v;
    }
#else
    if (e < LUT_N) {
        v4f v;
        v.x = lut[e];
        v.y = lut[LUT_N + e];
        v.z = lut[2 * LUT_N + e];
        v.w = 0.0f;
        lut4[e] = v;
    }
#endif
}

/* =========================================================================
 * Per-pixel trilinear blend.  8 corners -> 4 (g,b) cells, r-lerp inside a
 * cell; packed path: 2 aligned b128 gathers per cell (8 per pixel total).
 * ========================================================================= */
template <bool PACKED>
__device__ __forceinline__ void trilerp_px(float r, float g, float b,
                                           const v4f* __restrict__ lut4,
                                           const float* __restrict__ lutc,
                                           float& o0, float& o1, float& o2)
{
    const float inv = 32.0f / 1.000001f;  /* 1/binsize, constant-folded */
    const float rs = r * inv, gs = g * inv, bs = b * inv;
    const float rf = floorf(rs), gf = floorf(gs), bf = floorf(bs);
    const float rd = rs - rf,  gd = gs - gf,  bd = bs - bf;
    const int   base = (int)rf + (int)gf * LUT_DIM + (int)bf * (LUT_DIM * LUT_DIM);

    const float wg0 = 1.0f - gd, wb0 = 1.0f - bd, wr0 = 1.0f - rd;
    const float w[4]   = { wg0 * wb0, gd * wb0, wg0 * bd, gd * bd };
    const int   idx[4] = { base, base + LUT_DIM,
                           base + LUT_DIM * LUT_DIM,
                           base + LUT_DIM * LUT_DIM + LUT_DIM };

    float a0 = 0.0f, a1 = 0.0f, a2 = 0.0f;
#pragma unroll
    for (int c = 0; c < 4; ++c) {
        const float f0 = w[c] * wr0;
        const float f1 = w[c] * rd;
        if (PACKED) {
            const v4f c0 = lut4[idx[c]];
            const v4f c1 = lut4[idx[c] + 1];
            a0 = fmaf(f0, c0.x, fmaf(f1, c1.x, a0));
            a1 = fmaf(f0, c0.y, fmaf(f1, c1.y, a1));
            a2 = fmaf(f0, c0.z, fmaf(f1, c1.z, a2));
        } else {
            const int i0 = idx[c];
            a0 = fmaf(f0, lutc[i0],           fmaf(f1, lutc[i0 + 1],           a0));
            a1 = fmaf(f0, lutc[LUT_N + i0],   fmaf(f1, lutc[LUT_N + i0 + 1],   a1));
            a2 = fmaf(f0, lutc[2*LUT_N + i0], fmaf(f1, lutc[2*LUT_N + i0 + 1], a2));
        }
    }
    o0 = a0; o1 = a1; o2 = a2;
}

/* =========================================================================
 * Kernel B: streaming kernel.  One thread = 4 consecutive pixels of one
 * image: 3 NT b128 plane loads -> 8 b128 LUT gathers/pixel -> 3 NT b128
 * plane stores.  NT hints keep the 384 MB stream from evicting the 431 KB
 * L2-resident LUT.
 * ========================================================================= */
template <bool PACKED>
__global__ __launch_bounds__(256)
void trilerp_kernel(const float* __restrict__ x,
                    const v4f*  __restrict__ lut4,
                    const float* __restrict__ lutc,
                    float* __restrict__ out, int ngroups)
{
    const int i = blockIdx.x * blockDim.x + threadIdx.x;
    if (i >= ngroups) return;

    const int n  = i >> GRP_BITS;                 /* image index   */
    const int hw = i & ((1 << GRP_BITS) - 1);     /* float4 group  */

    const v4f* __restrict__ xin = (const v4f*)x;
    v4f* __restrict__       o   = (v4f*)out;
    const size_t pbase = ((size_t)n * 3) << GRP_BITS;

    const v4f r4 = __builtin_nontemporal_load(&xin[pbase + hw]);
    const v4f g4 = __builtin_nontemporal_load(&xin[pbase + (1 << GRP_BITS) + hw]);
    const v4f b4 = __builtin_nontemporal_load(&xin[pbase + (2 << GRP_BITS) + hw]);

    v4f o0, o1, o2;
#pragma unroll
    for (int k = 0; k < 4; ++k) {
        float c0, c1, c2;
        trilerp_px<PACKED>(r4[k], g4[k], b4[k], lut4, lutc, c0, c1, c2);
        o0[k] = c0; o1[k] = c1; o2[k] = c2;
    }

    __builtin_nontemporal_store(o0, &o[pbase + hw]);
    __builtin_nontemporal_store(o1, &o[pbase + (1 << GRP_BITS) + hw]);
    __builtin_nontemporal_store(o2, &o[pbase + (2 << GRP_BITS) + hw]);
}

extern "C" void kernel_launch(void* const* d_in, const int* in_sizes, int n_in,
                              void* d_out, int out_size, void* d_ws, size_t ws_size,
                              hipStream_t stream)
{
    const float* lut = (const float*)d_in[0];   /* (1,3,33,33,33) f32 */
    const float* x   = (const float*)d_in[1];   /* (16,3,1024,1024) f32 */
    float* out = (float*)d_out;

    const int npix    = in_sizes[1] / 3;        /* 16.7M pixels  */
    const int ngroups = npix / 4;               /* float4 groups */
    const int blocks  = (ngroups + 255) / 256;

    const size_t packed_bytes = (size_t)LUT_N * sizeof(v4f);  /* 575 KB */
    if (ws_size >= packed_bytes) {
        v4f* lut4 = (v4f*)d_ws;
        lut_pack_kernel<<<(LUT_N + 255) / 256, 256, 0, stream>>>(lut, lut4);
        trilerp_kernel<true><<<blocks, 256, 0, stream>>>(x, lut4, lut, out, ngroups);
    } else {
        /* scratch too small: channel-major gathers straight from the LUT */
        trilerp_kernel<false><<<blocks, 256, 0, stream>>>(x, nullptr, lut, out, ngroups);
    }
}